// LlamaAttention_23536420782102
// MI455X (gfx1250) — compile-verified
//
#include <hip/hip_runtime.h>
#include <hip/hip_bf16.h>

typedef __attribute__((ext_vector_type(2))) float v2f;
typedef __attribute__((ext_vector_type(8))) float v8f;
typedef unsigned int u32x4 __attribute__((ext_vector_type(4)));
typedef int          i32x8 __attribute__((ext_vector_type(8)));
typedef int          i32x4 __attribute__((ext_vector_type(4)));

#define B_    4
#define HID_  4096
#define H_    32
#define HKV_  8
#define G_    4
#define D_    128
#define S_    16384
#define PAGE_ 64
#define TOPK_ 64
#define NP_   256                 // pages per (b,h)
#define LTOT_ (TOPK_*PAGE_ + 1)   // 4097
#define SCALE_ 0.08838834764831845f

// ---------------------------------------------------------------------------
// 1) QKV projection: out[n] = dot(hidden[b], W[n,:]) for n over [q|k|v] rows.
// ---------------------------------------------------------------------------
__global__ void qkv_proj_kernel(const float* __restrict__ hid,
                                const float* __restrict__ Wq,
                                const float* __restrict__ Wk,
                                const float* __restrict__ Wv,
                                float* __restrict__ q_raw,
                                float* __restrict__ k_raw,
                                float* __restrict__ v_raw) {
  const int n = blockIdx.x;      // 0..6143
  const int t = threadIdx.x;     // 0..255
  const float* wrow;
  float* outp;
  int ostride;
  if (n < HID_)             { wrow = Wq + (size_t)n * HID_;             outp = q_raw + n;             ostride = HID_; }
  else if (n < HID_ + 1024) { wrow = Wk + (size_t)(n - HID_) * HID_;    outp = k_raw + (n - HID_);    ostride = 1024; }
  else                      { wrow = Wv + (size_t)(n - HID_ - 1024) * HID_; outp = v_raw + (n - HID_ - 1024); ostride = 1024; }

  float acc[B_] = {0.f, 0.f, 0.f, 0.f};
  const int base = t * 16;       // each thread: 16 contiguous floats (4 x float4)
  __builtin_prefetch(wrow + base, 0, 0);
  #pragma unroll
  for (int c = 0; c < 4; ++c) {
    const float4 w4 = *(const float4*)(wrow + base + c * 4);
    #pragma unroll
    for (int b = 0; b < B_; ++b) {
      const float4 x4 = *(const float4*)(hid + (size_t)b * HID_ + base + c * 4);
      acc[b] += w4.x * x4.x + w4.y * x4.y + w4.z * x4.z + w4.w * x4.w;
    }
  }
  __shared__ float red[256];
  #pragma unroll
  for (int b = 0; b < B_; ++b) {
    red[t] = acc[b];
    __syncthreads();
    for (int off = 128; off > 0; off >>= 1) {
      if (t < off) red[t] += red[t + off];
      __syncthreads();
    }
    if (t == 0) outp[(size_t)b * ostride] = red[0];
    __syncthreads();
  }
}

// ---------------------------------------------------------------------------
// 2) RoPE on q and k_new
// ---------------------------------------------------------------------------
__global__ void rope_kernel(const float* __restrict__ q_raw,
                            const float* __restrict__ k_raw,
                            const float* __restrict__ cosb,
                            const float* __restrict__ sinb,
                            float* __restrict__ q_rot,
                            float* __restrict__ k_rot) {
  const int idx = blockIdx.x;
  const int b   = idx / (H_ + HKV_);
  const int hh  = idx % (H_ + HKV_);
  const int d   = threadIdx.x;
  const float c = cosb[b * D_ + d];
  const float s = sinb[b * D_ + d];
  const float* src;
  float* dst;
  if (hh < H_) { src = q_raw + ((size_t)b * H_ + hh) * D_;        dst = q_rot + ((size_t)b * H_ + hh) * D_; }
  else         { src = k_raw + ((size_t)b * HKV_ + hh - H_) * D_; dst = k_rot + ((size_t)b * HKV_ + hh - H_) * D_; }
  const float x = src[d];
  const float r = (d < 64) ? -src[d + 64] : src[d - 64];
  dst[d] = x * c + r * s;
}

// ---------------------------------------------------------------------------
// 3) q_merged = mean over G query heads of roped q
// ---------------------------------------------------------------------------
__global__ void qmerge_kernel(const float* __restrict__ q_rot,
                              float* __restrict__ q_merged) {
  const int bh = blockIdx.x;
  const int b = bh / HKV_, h = bh % HKV_;
  const int d = threadIdx.x;
  float acc = 0.f;
  #pragma unroll
  for (int g = 0; g < G_; ++g)
    acc += q_rot[((size_t)b * H_ + h * G_ + g) * D_ + d];
  q_merged[(size_t)bh * D_ + d] = acc * 0.25f;
}

// ---------------------------------------------------------------------------
// 4) Page scoring with the Tensor Data Mover: one wave per page.
//    TDM loads the 64x128 f32 page tile (32KB) into LDS asynchronously,
//    wave waits on TENSORcnt, then reduces min/max per d from LDS.
// ---------------------------------------------------------------------------
__global__ void __launch_bounds__(32)
page_score_tdm_kernel(const float* __restrict__ kc,
                      const float* __restrict__ q_merged,
                      float* __restrict__ scores) {
  const int idx = blockIdx.x;            // bh*NP + p
  const int bh = idx / NP_, p = idx % NP_;
  const int lane = threadIdx.x;          // 0..31 (one wave)

  __shared__ float tile[PAGE_ * D_];     // 32 KB

  const float* src = kc + ((size_t)bh * S_ + (size_t)p * PAGE_) * D_;
  const unsigned long long gaddr = (unsigned long long)(uintptr_t)src;
  const unsigned int laddr = (unsigned int)(uintptr_t)(&tile[0]);

  // ---- D# group 0 (128b): count=1, lds_addr, global_addr(57b), type=2 ----
  u32x4 g0;
  g0.x = 1u;                                           // count=1, user mode
  g0.y = laddr;                                        // lds_addr (bytes)
  g0.z = (unsigned int)(gaddr & 0xFFFFFFFFull);        // global_addr[31:0]
  g0.w = (unsigned int)((gaddr >> 32) & 0x1FFFFFFull)  // global_addr[56:32]
         | (2u << 30);                                 // type=2 ("image")

  // ---- D# group 1 (256b): data_size=4B, dims/strides for 2D 128x64 tile --
  const unsigned int TD0 = D_;     // tensor_dim0 (elements)
  const unsigned int TD1 = PAGE_;  // tensor_dim1 (rows in this tile region)
  const unsigned int TL0 = D_;     // tile_dim0
  const unsigned int TL1 = PAGE_;  // tile_dim1
  const unsigned long long ST0 = D_;  // tensor_dim0_stride (elements)
  i32x8 g1;
  g1[0] = (int)(2u << 16);                              // wg_mask=0, data_size=2(4B)
  g1[1] = (int)((TD0 & 0xFFFFu) << 16);                 // tensor_dim0[15:0] @bits63:48
  g1[2] = (int)((TD0 >> 16) | ((TD1 & 0xFFFFu) << 16)); // dim0[31:16] | dim1[15:0]
  g1[3] = (int)((TD1 >> 16) | (TL0 << 16));             // dim1[31:16] | tile_dim0
  g1[4] = (int)(TL1 | (0u << 16));                      // tile_dim1 | tile_dim2=0
  g1[5] = (int)(unsigned int)(ST0 & 0xFFFFFFFFull);     // dim0_stride[31:0]
  g1[6] = (int)(unsigned int)(ST0 >> 32);               // dim0_stride[47:32] | dim1_stride[15:0]=0
  g1[7] = 0;                                            // dim1_stride[47:16]=0

  i32x4 g2 = {0, 0, 0, 0};                              // dims 2/3 unused
  i32x4 g3 = {0, 0, 0, 0};                              // dim 4 unused
  i32x8 gx = {0, 0, 0, 0, 0, 0, 0, 0};                  // trailing group (6-arg form)

  __builtin_amdgcn_tensor_load_to_lds(g0, g1, g2, g3, gx, 0);
  __builtin_amdgcn_s_wait_tensorcnt(0);

  // Each lane owns 4 consecutive d-columns: d = lane*4 + j
  float kmin[4] = { 3.4e38f,  3.4e38f,  3.4e38f,  3.4e38f};
  float kmax[4] = {-3.4e38f, -3.4e38f, -3.4e38f, -3.4e38f};
  #pragma unroll 4
  for (int row = 0; row < PAGE_; ++row) {
    const float4 v = *(const float4*)&tile[row * D_ + lane * 4];
    kmin[0] = fminf(kmin[0], v.x); kmax[0] = fmaxf(kmax[0], v.x);
    kmin[1] = fminf(kmin[1], v.y); kmax[1] = fmaxf(kmax[1], v.y);
    kmin[2] = fminf(kmin[2], v.z); kmax[2] = fmaxf(kmax[2], v.z);
    kmin[3] = fminf(kmin[3], v.w); kmax[3] = fmaxf(kmax[3], v.w);
  }
  const float4 qm = *(const float4*)&q_merged[(size_t)bh * D_ + lane * 4];
  float part = fmaxf(qm.x * kmin[0], qm.x * kmax[0])
             + fmaxf(qm.y * kmin[1], qm.y * kmax[1])
             + fmaxf(qm.z * kmin[2], qm.z * kmax[2])
             + fmaxf(qm.w * kmin[3], qm.w * kmax[3]);
  #pragma unroll
  for (int off = 16; off > 0; off >>= 1)
    part += __shfl_xor(part, off, 32);
  if (lane == 0) scores[idx] = part;
}

// ---------------------------------------------------------------------------
// 5) Top-64 of 256 page scores (lowest-index tie-break)
// ---------------------------------------------------------------------------
__global__ void topk_kernel(const float* __restrict__ scores,
                            int* __restrict__ topk) {
  const int bh = blockIdx.x;
  const int t = threadIdx.x;             // 0..255
  __shared__ float sv[256];
  __shared__ float rv[256];
  __shared__ int   ri[256];
  sv[t] = scores[bh * NP_ + t];
  __syncthreads();
  for (int i = 0; i < TOPK_; ++i) {
    rv[t] = sv[t]; ri[t] = t;
    __syncthreads();
    for (int off = 128; off > 0; off >>= 1) {
      if (t < off) {
        const float v1 = rv[t + off];
        if (v1 > rv[t] || (v1 == rv[t] && ri[t + off] < ri[t])) { rv[t] = v1; ri[t] = ri[t + off]; }
      }
      __syncthreads();
    }
    if (t == 0) { topk[bh * TOPK_ + i] = ri[0]; sv[ri[0]] = -3.4e38f; }
    __syncthreads();
  }
}

// ---------------------------------------------------------------------------
// 6) Q.K^T via V_WMMA_F32_16X16X4_F32, branch-free operand fetch:
//    all lanes load from a valid q row (lm&3) and multiply by a 0/1 mask,
//    keeping EXEC all-1s with no saveexec churn in the hot loop.
// ---------------------------------------------------------------------------
__global__ void __launch_bounds__(32)
qk_wmma_kernel(const float* __restrict__ kc,
               const float* __restrict__ q_rot,
               const int* __restrict__ topk,
               float* __restrict__ logits) {
  const int blk  = blockIdx.x;          // bh*256 + tile
  const int bh   = blk >> 8;
  const int tile = blk & 255;
  const int b = bh / HKV_, h = bh % HKV_;
  const int lane = threadIdx.x;         // 0..31 (one wave)
  const int half = lane >> 4;
  const int lm   = lane & 15;

  // B operand: this lane's token column (gathered)
  const int tl   = tile * 16 + lm;
  const int page = topk[bh * TOPK_ + (tl >> 6)];
  const int tok  = page * PAGE_ + (tl & 63);
  const float* krow = kc + ((size_t)bh * S_ + tok) * D_;
  // A operand: always-valid row, masked to zero for lm >= G
  const float asel = (lm < G_) ? 1.f : 0.f;
  const float* qrow = q_rot + ((size_t)b * H_ + h * G_ + (lm & 3)) * D_;

  v8f c = {};
  #pragma unroll 4
  for (int kk = 0; kk < 32; ++kk) {
    const int k0 = kk * 4 + half * 2;
    v2f a  = *(const v2f*)(qrow + k0);
    v2f bb = *(const v2f*)(krow + k0);
    a.x *= asel;
    a.y *= asel;
    c = __builtin_amdgcn_wmma_f32_16x16x4_f32(false, a, false, bb, (short)0, c, false, false);
  }
  #pragma unroll
  for (int r = 0; r < 8; ++r) {
    const int M = r + half * 8;          // C layout: vgpr r -> row M
    if (M < G_)
      logits[((size_t)bh * G_ + M) * LTOT_ + tl] = c[r] * SCALE_;
  }
}

// ---------------------------------------------------------------------------
// 6b) Logit for the appended new token (l = 4096)
// ---------------------------------------------------------------------------
__global__ void logit_last_kernel(const float* __restrict__ q_rot,
                                  const float* __restrict__ k_rot,
                                  float* __restrict__ logits) {
  const int bh = blockIdx.x;
  const int t = threadIdx.x;             // 0..127
  const int b = bh / HKV_, h = bh % HKV_;
  __shared__ float red[128];
  const float kv = k_rot[(size_t)bh * D_ + t];
  for (int g = 0; g < G_; ++g) {
    red[t] = q_rot[((size_t)b * H_ + h * G_ + g) * D_ + t] * kv;
    __syncthreads();
    for (int off = 64; off > 0; off >>= 1) {
      if (t < off) red[t] += red[t + off];
      __syncthreads();
    }
    if (t == 0) logits[((size_t)bh * G_ + g) * LTOT_ + (LTOT_ - 1)] = red[0] * SCALE_;
    __syncthreads();
  }
}

// ---------------------------------------------------------------------------
// 7) Softmax statistics (max, sum of exp) per (b,h,g)
// ---------------------------------------------------------------------------
__global__ void softmax_stats_kernel(const float* __restrict__ logits,
                                     float* __restrict__ stats) {
  const int i = blockIdx.x;              // 0..127
  const int t = threadIdx.x;             // 0..255
  const float* lp = logits + (size_t)i * LTOT_;
  float mx = -3.4e38f;
  for (int l = t; l < LTOT_; l += 256) mx = fmaxf(mx, lp[l]);
  __shared__ float red[256];
  red[t] = mx;
  __syncthreads();
  for (int off = 128; off > 0; off >>= 1) {
    if (t < off) red[t] = fmaxf(red[t], red[t + off]);
    __syncthreads();
  }
  const float m = red[0];
  __syncthreads();
  float sm = 0.f;
  for (int l = t; l < LTOT_; l += 256) sm += __expf(lp[l] - m);
  red[t] = sm;
  __syncthreads();
  for (int off = 128; off > 0; off >>= 1) {
    if (t < off) red[t] += red[t + off];
    __syncthreads();
  }
  if (t == 0) { stats[2 * i] = m; stats[2 * i + 1] = red[0]; }
}

// ---------------------------------------------------------------------------
// 8) o = softmax(w) @ V via WMMA; branch-free 1024-step main loop over the
//    4096 gathered tokens, peeled tail for the appended token l = 4096.
// ---------------------------------------------------------------------------
__global__ void __launch_bounds__(32)
pv_wmma_kernel(const float* __restrict__ vc,
               const float* __restrict__ v_raw,
               const float* __restrict__ logits,
               const float* __restrict__ stats,
               const int* __restrict__ topk,
               float* __restrict__ o_heads) {
  const int blk = blockIdx.x;            // bh*8 + dchunk
  const int bh  = blk >> 3;
  const int d0  = (blk & 7) * 16;
  const int b = bh / HKV_, h = bh % HKV_;
  const int lane = threadIdx.x;          // 0..31
  const int half = lane >> 4;
  const int lm   = lane & 15;

  const float asel = (lm < G_) ? 1.f : 0.f;
  const int grow = bh * G_ + (lm & 3);                 // always-valid row
  const float* lp = logits + (size_t)grow * LTOT_;
  const float  m  = stats[2 * grow];
  const float  ws = asel / stats[2 * grow + 1];        // mask folded into 1/sum

  const int* tkp = topk + bh * TOPK_;
  const float* vbase = vc + (size_t)bh * S_ * D_;
  const float* vnew  = v_raw + (size_t)bh * D_;
  const int dcol = d0 + lm;

  v8f c = {};
  for (int kq = 0; kq < TOPK_ * PAGE_ / 4; ++kq) {     // 1024 branch-free steps
    const int l0 = kq * 4 + half * 2;
    const int l1 = l0 + 1;
    const int tok0 = tkp[l0 >> 6] * PAGE_ + (l0 & 63);
    const int tok1 = tkp[l1 >> 6] * PAGE_ + (l1 & 63);
    v2f a, bb;
    a.x  = __expf(lp[l0] - m) * ws;
    a.y  = __expf(lp[l1] - m) * ws;
    bb.x = vbase[(size_t)tok0 * D_ + dcol];
    bb.y = vbase[(size_t)tok1 * D_ + dcol];
    c = __builtin_amdgcn_wmma_f32_16x16x4_f32(false, a, false, bb, (short)0, c, false, false);
  }
  // tail: l = 4096 (new token); only half==0 lanes carry K-slot 4096
  {
    v2f a, bb;
    const float wlast = (half == 0) ? __expf(lp[LTOT_ - 1] - m) * ws : 0.f;
    const float vlast = (half == 0) ? vnew[dcol] : 0.f;
    a.x = wlast; a.y = 0.f;
    bb.x = vlast; bb.y = 0.f;
    c = __builtin_amdgcn_wmma_f32_16x16x4_f32(false, a, false, bb, (short)0, c, false, false);
  }
  #pragma unroll
  for (int r = 0; r < 8; ++r) {
    const int M = r + half * 8;
    if (M < G_)
      o_heads[((size_t)b * H_ + h * G_ + M) * D_ + d0 + lm] = c[r];
  }
}

// ---------------------------------------------------------------------------
// 9) Output projection: out[b,j] = dot(o_flat[b], Wo[j,:])
// ---------------------------------------------------------------------------
__global__ void out_proj_kernel(const float* __restrict__ o_heads,
                                const float* __restrict__ Wo,
                                float* __restrict__ out) {
  const int j = blockIdx.x;              // 0..4095
  const int t = threadIdx.x;             // 0..255
  const float* wrow = Wo + (size_t)j * HID_;
  float acc[B_] = {0.f, 0.f, 0.f, 0.f};
  const int base = t * 16;
  #pragma unroll
  for (int c = 0; c < 4; ++c) {
    const float4 w4 = *(const float4*)(wrow + base + c * 4);
    #pragma unroll
    for (int b = 0; b < B_; ++b) {
      const float4 x4 = *(const float4*)(o_heads + (size_t)b * HID_ + base + c * 4);
      acc[b] += w4.x * x4.x + w4.y * x4.y + w4.z * x4.z + w4.w * x4.w;
    }
  }
  __shared__ float red[256];
  #pragma unroll
  for (int b = 0; b < B_; ++b) {
    red[t] = acc[b];
    __syncthreads();
    for (int off = 128; off > 0; off >>= 1) {
      if (t < off) red[t] += red[t + off];
      __syncthreads();
    }
    if (t == 0) out[(size_t)b * HID_ + j] = red[0];
    __syncthreads();
  }
}

// ---------------------------------------------------------------------------
extern "C" void kernel_launch(void* const* d_in, const int* in_sizes, int n_in,
                              void* d_out, int out_size, void* d_ws, size_t ws_size,
                              hipStream_t stream) {
  (void)in_sizes; (void)n_in; (void)out_size; (void)ws_size;
  const float* hid  = (const float*)d_in[0];
  const float* cosb = (const float*)d_in[1];
  const float* sinb = (const float*)d_in[2];
  const float* kc   = (const float*)d_in[3];
  const float* vc   = (const float*)d_in[4];
  const float* Wq   = (const float*)d_in[5];
  const float* Wk   = (const float*)d_in[6];
  const float* Wv   = (const float*)d_in[7];
  const float* Wo   = (const float*)d_in[8];
  float* out = (float*)d_out;

  float* ws = (float*)d_ws;
  float* q_raw    = ws; ws += B_ * H_ * D_;                 // 16384
  float* k_raw    = ws; ws += B_ * HKV_ * D_;               // 4096
  float* v_raw    = ws; ws += B_ * HKV_ * D_;               // 4096
  float* q_rot    = ws; ws += B_ * H_ * D_;                 // 16384
  float* k_rot    = ws; ws += B_ * HKV_ * D_;               // 4096
  float* q_merged = ws; ws += B_ * HKV_ * D_;               // 4096
  float* scores   = ws; ws += B_ * HKV_ * NP_;              // 8192
  int*   topk     = (int*)ws; ws += B_ * HKV_ * TOPK_;      // 2048
  float* logits   = ws; ws += (size_t)B_ * HKV_ * G_ * LTOT_; // 524416
  float* stats    = ws; ws += B_ * HKV_ * G_ * 2;           // 256
  float* o_heads  = ws; ws += B_ * H_ * D_;                 // 16384

  qkv_proj_kernel<<<HID_ + 2 * HKV_ * D_, 256, 0, stream>>>(hid, Wq, Wk, Wv, q_raw, k_raw, v_raw);
  rope_kernel<<<B_ * (H_ + HKV_), D_, 0, stream>>>(q_raw, k_raw, cosb, sinb, q_rot, k_rot);
  qmerge_kernel<<<B_ * HKV_, D_, 0, stream>>>(q_rot, q_merged);
  page_score_tdm_kernel<<<B_ * HKV_ * NP_, 32, 0, stream>>>(kc, q_merged, scores);
  topk_kernel<<<B_ * HKV_, NP_, 0, stream>>>(scores, topk);
  qk_wmma_kernel<<<B_ * HKV_ * (TOPK_ * PAGE_ / 16), 32, 0, stream>>>(kc, q_rot, topk, logits);
  logit_last_kernel<<<B_ * HKV_, D_, 0, stream>>>(q_rot, k_rot, logits);
  softmax_stats_kernel<<<B_ * HKV_ * G_, 256, 0, stream>>>(logits, stats);
  pv_wmma_kernel<<<B_ * HKV_ * (D_ / 16), 32, 0, stream>>>(vc, v_raw, logits, stats, topk, o_heads);
  out_proj_kernel<<<HID_, 256, 0, stream>>>(o_heads, Wo, out);
}